// GATModel_87419764342917
// MI455X (gfx1250) — compile-verified
//
#include <hip/hip_runtime.h>

// ---------------- problem constants (match reference) ----------------
#define NN   100000
#define EE   600000
#define ETOT (EE + NN)     // + self loops
#define INF_ 64
#define HF   128
#define GG   5000
#define OUTF 19
#define NEG_SLOPE 0.2f

typedef __attribute__((ext_vector_type(2))) float v2f;
typedef __attribute__((ext_vector_type(8))) float v8f;

// monotone order-preserving key for float -> uint (for atomicMax segment-max)
__device__ __forceinline__ unsigned fkey(float f) {
    unsigned u = __float_as_uint(f);
    return (u & 0x80000000u) ? ~u : (u | 0x80000000u);
}
__device__ __forceinline__ float funkey(unsigned k) {
    return (k & 0x80000000u) ? __uint_as_float(k ^ 0x80000000u) : __uint_as_float(~k);
}

// ---------------------------------------------------------------------
// GEMM: h[N,128] = in[N,K] @ W[K,128] using V_WMMA_F32_16X16X4_F32.
// One wave computes a 16-row x 128-col strip (8 accumulators of 16x16).
// Also fuses s_src = h @ a_src, s_dst = h @ a_dst per row.
// A layout (16x4 f32): lane&15 = M, lanes>=16 hold K+2; VGPR0/1 = K, K+1.
// B layout (4x16 f32): lane&15 = N, lanes>=16 hold K+2; VGPR0/1 = K, K+1.
// C/D layout: VGPR j = row j (lanes 0-15) / row 8+j (lanes 16-31), lane&15 = col.
// ---------------------------------------------------------------------
__global__ void __launch_bounds__(256)
gemm_wmma(const float* __restrict__ in, const float* __restrict__ W,
          const float* __restrict__ a_src, const float* __restrict__ a_dst,
          float* __restrict__ h, float* __restrict__ s_src, float* __restrict__ s_dst,
          int K)
{
    const int wave = threadIdx.x >> 5;
    const int lane = threadIdx.x & 31;
    const int tile = blockIdx.x * 8 + wave;           // 16-row tile
    if (tile * 16 >= NN) return;                      // wave-uniform: EXEC stays full
    const int row0 = tile * 16;
    const int r    = lane & 15;
    const int hi   = lane >> 4;                       // 0 or 1
    const int koff = hi * 2;

    v8f acc[8];
    const v8f vz = {0.f,0.f,0.f,0.f,0.f,0.f,0.f,0.f};
#pragma unroll
    for (int t = 0; t < 8; ++t) acc[t] = vz;

    const float* arow = in + (size_t)(row0 + r) * K;
    for (int k0 = 0; k0 < K; k0 += 4) {
        v2f a;
        a.x = arow[k0 + koff];
        a.y = arow[k0 + koff + 1];
        const float* wrow = W + (size_t)(k0 + koff) * HF + r;
#pragma unroll
        for (int t = 0; t < 8; ++t) {
            v2f b;
            b.x = wrow[t * 16];
            b.y = wrow[HF + t * 16];
            acc[t] = __builtin_amdgcn_wmma_f32_16x16x4_f32(
                false, a, false, b, (short)0, acc[t], false, false);
        }
    }

    // store h and fused per-row dot products with a_src / a_dst
    float ps[8], pd[8];
#pragma unroll
    for (int j = 0; j < 8; ++j) { ps[j] = 0.f; pd[j] = 0.f; }
#pragma unroll
    for (int t = 0; t < 8; ++t) {
        const int col = t * 16 + r;
        const float as = a_src[col];
        const float ad = a_dst[col];
#pragma unroll
        for (int j = 0; j < 8; ++j) {
            const int row = row0 + hi * 8 + j;
            float v = acc[t][j];
            h[(size_t)row * HF + col] = v;
            ps[j] += v * as;
            pd[j] += v * ad;
        }
    }
    // reduce across the 16 lanes of each half-wave (masks 1,2,4,8 stay in-half)
#pragma unroll
    for (int off = 1; off < 16; off <<= 1) {
#pragma unroll
        for (int j = 0; j < 8; ++j) {
            ps[j] += __shfl_xor(ps[j], off, 32);
            pd[j] += __shfl_xor(pd[j], off, 32);
        }
    }
    if (r == 0) {
#pragma unroll
        for (int j = 0; j < 8; ++j) {
            const int row = row0 + hi * 8 + j;
            s_src[row] = ps[j];
            s_dst[row] = pd[j];
        }
    }
}

// ---------------- per-layer init: acc=0, m=key(-inf)=0, denom=0 -------
__global__ void layer_init(float* __restrict__ acc, unsigned* __restrict__ m,
                           float* __restrict__ denom)
{
    int i = blockIdx.x * blockDim.x + threadIdx.x;
    if (i < NN * (HF / 4)) {
        float4 z = {0.f, 0.f, 0.f, 0.f};
        ((float4*)acc)[i] = z;
    }
    if (i < NN) { m[i] = 0u; denom[i] = 0.f; }
}

// ---------- pass 1: alpha = leakyrelu(s_src[src]+s_dst[dst]); segmax --
__global__ void edge_alpha_max(const int* __restrict__ esrc, const int* __restrict__ edst,
                               const float* __restrict__ s_src, const float* __restrict__ s_dst,
                               float* __restrict__ alpha, unsigned* __restrict__ m)
{
    int e = blockIdx.x * blockDim.x + threadIdx.x;
    if (e >= ETOT) return;
    int s = (e < EE) ? esrc[e] : (e - EE);
    int d = (e < EE) ? edst[e] : (e - EE);
    float a = s_src[s] + s_dst[d];
    a = (a > 0.f) ? a : NEG_SLOPE * a;
    alpha[e] = a;
    atomicMax(m + d, fkey(a));
}

// ---------- pass 2: e = exp(alpha - m[dst]); denom[dst] += e ----------
__global__ void edge_exp_sum(const int* __restrict__ edst,
                             float* __restrict__ alpha, const unsigned* __restrict__ m,
                             float* __restrict__ denom)
{
    int e = blockIdx.x * blockDim.x + threadIdx.x;
    if (e >= ETOT) return;
    int d = (e < EE) ? edst[e] : (e - EE);
    float ex = expf(alpha[e] - funkey(m[d]));
    alpha[e] = ex;
    atomicAdd(denom + d, ex);
}

// ---------- pass 3: acc[dst,:] += (e/denom[dst]) * h[src,:] -----------
// one wave per edge; each lane handles 4 contiguous floats
__global__ void edge_scatter(const int* __restrict__ esrc, const int* __restrict__ edst,
                             const float* __restrict__ h, const float* __restrict__ alpha,
                             const float* __restrict__ denom, float* __restrict__ acc)
{
    int gw = (int)((blockIdx.x * (size_t)blockDim.x + threadIdx.x) >> 5);
    int lane = threadIdx.x & 31;
    if (gw >= ETOT) return;
    int s = (gw < EE) ? esrc[gw] : (gw - EE);
    int d = (gw < EE) ? edst[gw] : (gw - EE);
    float coef = alpha[gw] / denom[d];
    float4 hv = ((const float4*)(h + (size_t)s * HF))[lane];
    float* ap = acc + (size_t)d * HF + lane * 4;
    atomicAdd(ap + 0, coef * hv.x);
    atomicAdd(ap + 1, coef * hv.y);
    atomicAdd(ap + 2, coef * hv.z);
    atomicAdd(ap + 3, coef * hv.w);
}

// ---------------- act = relu(acc + bias) ------------------------------
__global__ void bias_relu(const float* __restrict__ acc, const float* __restrict__ b,
                          float* __restrict__ act)
{
    int i = blockIdx.x * blockDim.x + threadIdx.x;
    if (i >= NN * HF) return;
    float v = acc[i] + b[i & (HF - 1)];
    act[i] = (v > 0.f) ? v : 0.f;
}

// ---------------- pooling -------------------------------------------
__global__ void pool_init(float* __restrict__ pooled, float* __restrict__ cnts)
{
    int i = blockIdx.x * blockDim.x + threadIdx.x;
    if (i < GG * HF) pooled[i] = 0.f;
    if (i < GG) cnts[i] = 0.f;
}

__global__ void pool_sum(const float* __restrict__ act, const int* __restrict__ batch,
                         float* __restrict__ pooled, float* __restrict__ cnts)
{
    int t = blockIdx.x * blockDim.x + threadIdx.x;
    if (t >= NN * (HF / 4)) return;
    int node = t >> 5, c = t & 31;
    int b = batch[node];
    float4 v = ((const float4*)(act + (size_t)node * HF))[c];
    float* p = pooled + (size_t)b * HF + c * 4;
    atomicAdd(p + 0, v.x);
    atomicAdd(p + 1, v.y);
    atomicAdd(p + 2, v.z);
    atomicAdd(p + 3, v.w);
    if (c == 0) atomicAdd(cnts + b, 1.0f);
}

// ---------------- FC head: relu(pooled@fc1+b1)@fc2+b2 ----------------
__global__ void __launch_bounds__(64)
fc_head(const float* __restrict__ pooled, const float* __restrict__ cnts,
        const float* __restrict__ w1, const float* __restrict__ b1,
        const float* __restrict__ w2, const float* __restrict__ b2,
        float* __restrict__ out)
{
    __shared__ float p[HF];
    __shared__ float z[64];
    int g = blockIdx.x, t = threadIdx.x;
    float inv = 1.0f / fmaxf(cnts[g], 1.0f);
    p[t]      = pooled[(size_t)g * HF + t] * inv;
    p[t + 64] = pooled[(size_t)g * HF + t + 64] * inv;
    __syncthreads();
    float a = b1[t];
    for (int k = 0; k < HF; ++k) a += p[k] * w1[k * 64 + t];
    z[t] = fmaxf(a, 0.f);
    __syncthreads();
    if (t < OUTF) {
        float o = b2[t];
        for (int k = 0; k < 64; ++k) o += z[k] * w2[k * OUTF + t];
        out[(size_t)g * OUTF + t] = o;
    }
}

// ---------------------------------------------------------------------
extern "C" void kernel_launch(void* const* d_in, const int* in_sizes, int n_in,
                              void* d_out, int out_size, void* d_ws, size_t ws_size,
                              hipStream_t stream)
{
    const float* x      = (const float*)d_in[0];
    const int*   eidx   = (const int*)  d_in[1];
    const int*   batch  = (const int*)  d_in[2];
    const float* W[3]    = {(const float*)d_in[3],  (const float*)d_in[7],  (const float*)d_in[11]};
    const float* asrc[3] = {(const float*)d_in[4],  (const float*)d_in[8],  (const float*)d_in[12]};
    const float* adst[3] = {(const float*)d_in[5],  (const float*)d_in[9],  (const float*)d_in[13]};
    const float* bias[3] = {(const float*)d_in[6],  (const float*)d_in[10], (const float*)d_in[14]};
    const float* fc1w = (const float*)d_in[15];
    const float* fc1b = (const float*)d_in[16];
    const float* fc2w = (const float*)d_in[17];
    const float* fc2b = (const float*)d_in[18];
    float* out = (float*)d_out;

    const int* esrc = eidx;        // edge_index[0,:]
    const int* edst = eidx + EE;   // edge_index[1,:]

    // ---- carve workspace (<= ~161 MB) ----
    char* ws = (char*)d_ws;
    const size_t NHB = (size_t)NN * HF * sizeof(float);          // 51.2 MB
    float*    h_lin  = (float*)(ws);
    float*    acc    = (float*)(ws + NHB);
    float*    act    = (float*)(ws + 2 * NHB);
    float*    s_src  = (float*)(ws + 3 * NHB);
    float*    s_dst  = (float*)(ws + 3 * NHB + (size_t)NN * 4);
    unsigned* m      = (unsigned*)(ws + 3 * NHB + (size_t)NN * 8);
    float*    denom  = (float*)(ws + 3 * NHB + (size_t)NN * 12);
    float*    alpha  = (float*)(ws + 3 * NHB + (size_t)NN * 16);
    float*    pooled = (float*)(ws + 3 * NHB + (size_t)NN * 16 + (size_t)ETOT * 4);
    float*    cnts   = (float*)(ws + 3 * NHB + (size_t)NN * 16 + (size_t)ETOT * 4 + (size_t)GG * HF * 4);

    const int gemm_grid   = ((NN / 16) + 7) / 8;                 // 782 blocks of 8 waves
    const int init_grid   = (NN * (HF / 4) + 255) / 256;
    const int edge_grid   = (ETOT + 255) / 256;
    const int scat_grid   = (int)(((size_t)ETOT * 32 + 255) / 256);
    const int br_grid     = (NN * HF + 255) / 256;
    const int pool_grid   = (NN * (HF / 4) + 255) / 256;
    const int pinit_grid  = (GG * HF + 255) / 256;

    const float* cur = x;
    int K = INF_;
    for (int l = 0; l < 3; ++l) {
        gemm_wmma<<<gemm_grid, 256, 0, stream>>>(cur, W[l], asrc[l], adst[l],
                                                 h_lin, s_src, s_dst, K);
        layer_init<<<init_grid, 256, 0, stream>>>(acc, m, denom);
        edge_alpha_max<<<edge_grid, 256, 0, stream>>>(esrc, edst, s_src, s_dst, alpha, m);
        edge_exp_sum<<<edge_grid, 256, 0, stream>>>(edst, alpha, m, denom);
        edge_scatter<<<scat_grid, 256, 0, stream>>>(esrc, edst, h_lin, alpha, denom, acc);
        bias_relu<<<br_grid, 256, 0, stream>>>(acc, bias[l], act);
        cur = act;
        K = HF;
    }

    pool_init<<<pinit_grid, 256, 0, stream>>>(pooled, cnts);
    pool_sum<<<pool_grid, 256, 0, stream>>>(act, batch, pooled, cnts);
    fc_head<<<GG, 64, 0, stream>>>(pooled, cnts, fc1w, fc1b, fc2w, fc2b, out);
}